// GraphConvolution_pynq_69664369541618
// MI455X (gfx1250) — compile-verified
//
#include <hip/hip_runtime.h>

typedef float v2f __attribute__((ext_vector_type(2)));
typedef float v8f __attribute__((ext_vector_type(8)));
typedef unsigned int u32x4 __attribute__((ext_vector_type(4)));
typedef int i32x8 __attribute__((ext_vector_type(8)));
typedef int i32x4 __attribute__((ext_vector_type(4)));

#define F_IN  512
#define F_OUT 256
#define KCH   64     // K-chunk staged in LDS per iteration
#define EPB   256    // edges per block in spmm

// ---------------------------------------------------------------------------
// Zero the output buffer (harness poisons d_out with 0xAA before timing).
// ---------------------------------------------------------------------------
__global__ __launch_bounds__(256)
void zero_f4(float4* __restrict__ p, long n4) {
    long i = (long)blockIdx.x * blockDim.x + threadIdx.x;
    if (i < n4) p[i] = float4{0.f, 0.f, 0.f, 0.f};
}

// ---------------------------------------------------------------------------
// TDM: issue a 2D tensor_load_to_lds (Tensor Data Mover DMA, CDNA5 §8).
// D# group0: count=1 | lds_addr | global_addr | type=2 ("image").
// D# group1: data_size=4B, tensor_dim0/1, tile_dim0/1, dim0 row stride.
// Groups 2/3 zeroed (<=2D tensor). Tracked with TENSORcnt.
// clang-23 form: 6 args (g0 u32x4, g1 i32x8, g2 i32x4, g3 i32x4, i32x8, cpol).
// ---------------------------------------------------------------------------
__device__ __forceinline__
void tdm_load_2d(unsigned lds_off, const void* gptr,
                 unsigned tensor_d0, unsigned tensor_d1,
                 unsigned tile_d0, unsigned tile_d1,
                 unsigned long long row_stride_elts) {
    unsigned long long ga = (unsigned long long)(uintptr_t)gptr;
    u32x4 g0;
    g0[0] = 1u;                                            // count=1, user desc
    g0[1] = lds_off;                                       // LDS byte address
    g0[2] = (unsigned)(ga & 0xffffffffull);                // global addr lo
    g0[3] = (unsigned)((ga >> 32) & 0x1ffffffull)          // global addr hi
            | (2u << 30);                                  // type = 2
    i32x8 g1;
    g1[0] = (int)(2u << 16);                               // data_size = 4 bytes
    g1[1] = (int)((tensor_d0 & 0xffffu) << 16);            // tensor_dim0[15:0]
    g1[2] = (int)((tensor_d0 >> 16) | ((tensor_d1 & 0xffffu) << 16));
    g1[3] = (int)((tensor_d1 >> 16) | (tile_d0 << 16));    // tile_dim0
    g1[4] = (int)(tile_d1 & 0xffffu);                      // tile_dim1 (tile_dim2=0)
    g1[5] = (int)(row_stride_elts & 0xffffffffull);        // dim0 stride lo32
    g1[6] = (int)((row_stride_elts >> 32) & 0xffffull);    // dim0 stride hi16
    g1[7] = 0;
    i32x4 z4 = {0, 0, 0, 0};
    i32x8 z8 = {0, 0, 0, 0, 0, 0, 0, 0};
    __builtin_amdgcn_tensor_load_to_lds(g0, g1, z4, z4, z8, 0);
}

// ---------------------------------------------------------------------------
// Dense GEMM: support[n, F_OUT] = x[n, F_IN] @ w[F_IN, F_OUT]  (fp32 WMMA)
//
// Block: 256 threads = 8 waves; block tile 64(M) x 256(N).
// Wave tile: 32(M) x 64(N) -> 8 accumulators of V_WMMA_F32_16X16X4_F32
// (B fragments reused across 2 M-tiles).
// K staged in 8 double-buffered LDS chunks of 64, filled by the TDM
// (tensor_load_to_lds), synchronized with s_wait_tensorcnt + barrier.
// Fragments read from LDS: A = ds_load_b64, B = 2x ds_load_b32.
// ---------------------------------------------------------------------------
__global__ __launch_bounds__(256)
void gemm_xw_wmma(const float* __restrict__ x, const float* __restrict__ w,
                  float* __restrict__ support, int n_nodes) {
    __shared__ float lds_x[2][64 * KCH];    // 2 x 16 KB  : x tile (64 rows x KCH)
    __shared__ float lds_w[2][KCH * F_OUT]; // 2 x 64 KB  : w tile (KCH rows x 256)

    const int lane  = threadIdx.x & 31;
    const int wave  = threadIdx.x >> 5;    // 0..7
    const int waveM = wave >> 2;           // 0..1 -> 32 rows each
    const int waveN = wave & 3;            // 0..3 -> 64 cols each
    const int m_base = blockIdx.x * 64;
    const int n0 = waveN * 64;
    const int lm = lane & 15;              // 0..15
    const int kg = lane >> 4;              // 0/1: K half of the fragment

    unsigned xrows = (unsigned)(n_nodes - m_base);
    if (xrows > 64u) xrows = 64u;

    // ---- prologue: DMA chunk 0 (x tile + w tile) ----
    if (threadIdx.x == 0) {
        tdm_load_2d((unsigned)(uintptr_t)&lds_x[0][0], x + (long)m_base * F_IN,
                    KCH, xrows, KCH, 64, F_IN);
        tdm_load_2d((unsigned)(uintptr_t)&lds_w[0][0], w,
                    F_OUT, KCH, F_OUT, KCH, F_OUT);
        __builtin_amdgcn_s_wait_tensorcnt(0);
    }
    __syncthreads();

    v8f acc00 = {}, acc01 = {}, acc02 = {}, acc03 = {};   // M-tile 0, N-tiles 0..3
    v8f acc10 = {}, acc11 = {}, acc12 = {}, acc13 = {};   // M-tile 1, N-tiles 0..3

    const int NCHUNKS = F_IN / KCH;        // 8
    for (int c = 0; c < NCHUNKS; ++c) {
        const int buf = c & 1;

        // ---- async: DMA next chunk into the other buffer while computing ----
        if (threadIdx.x == 0 && (c + 1) < NCHUNKS) {
            const int kn = (c + 1) * KCH;
            tdm_load_2d((unsigned)(uintptr_t)&lds_x[buf ^ 1][0],
                        x + (long)m_base * F_IN + kn,
                        KCH, xrows, KCH, 64, F_IN);
            tdm_load_2d((unsigned)(uintptr_t)&lds_w[buf ^ 1][0],
                        w + (long)kn * F_OUT,
                        F_OUT, KCH, F_OUT, KCH, F_OUT);
        }

        const float* __restrict__ xb = &lds_x[buf][0];
        const float* __restrict__ wb = &lds_w[buf][0];
        const float* __restrict__ xa0 = xb + (waveM * 32 + lm) * KCH + 2 * kg;
        const float* __restrict__ xa1 = xa0 + 16 * KCH;
        const float* __restrict__ wc  = wb + 2 * kg * F_OUT + n0 + lm;

#pragma unroll 4
        for (int k0 = 0; k0 < KCH; k0 += 4) {
            v2f a0 = *(const v2f*)(xa0 + k0);            // ds_load_b64
            v2f a1 = *(const v2f*)(xa1 + k0);
            const float* __restrict__ wr = wc + k0 * F_OUT;
            v2f b0 = { wr[ 0], wr[F_OUT +  0] };         // 2x ds_load_b32
            v2f b1 = { wr[16], wr[F_OUT + 16] };
            v2f b2 = { wr[32], wr[F_OUT + 32] };
            v2f b3 = { wr[48], wr[F_OUT + 48] };
            acc00 = __builtin_amdgcn_wmma_f32_16x16x4_f32(false, a0, false, b0, (short)0, acc00, false, false);
            acc01 = __builtin_amdgcn_wmma_f32_16x16x4_f32(false, a0, false, b1, (short)0, acc01, false, false);
            acc02 = __builtin_amdgcn_wmma_f32_16x16x4_f32(false, a0, false, b2, (short)0, acc02, false, false);
            acc03 = __builtin_amdgcn_wmma_f32_16x16x4_f32(false, a0, false, b3, (short)0, acc03, false, false);
            acc10 = __builtin_amdgcn_wmma_f32_16x16x4_f32(false, a1, false, b0, (short)0, acc10, false, false);
            acc11 = __builtin_amdgcn_wmma_f32_16x16x4_f32(false, a1, false, b1, (short)0, acc11, false, false);
            acc12 = __builtin_amdgcn_wmma_f32_16x16x4_f32(false, a1, false, b2, (short)0, acc12, false, false);
            acc13 = __builtin_amdgcn_wmma_f32_16x16x4_f32(false, a1, false, b3, (short)0, acc13, false, false);
        }

        // DMA for chunk c+1 must land before the next compute iteration.
        if (threadIdx.x == 0) __builtin_amdgcn_s_wait_tensorcnt(0);
        __syncthreads();
    }

    // ---- store D tiles ----
    // C/D layout: VGPR r, lane half kg -> row (+8*kg + r), col = n0 + 16*tile + lm
    const int crow0 = m_base + waveM * 32 + kg * 8;        // M-tile 0
    const int crow1 = crow0 + 16;                          // M-tile 1
    float* __restrict__ op0 = support + (long)crow0 * F_OUT + n0 + lm;
    float* __restrict__ op1 = support + (long)crow1 * F_OUT + n0 + lm;
#pragma unroll
    for (int r = 0; r < 8; ++r) {
        if (crow0 + r < n_nodes) {
            op0[(long)r * F_OUT +  0] = acc00[r];
            op0[(long)r * F_OUT + 16] = acc01[r];
            op0[(long)r * F_OUT + 32] = acc02[r];
            op0[(long)r * F_OUT + 48] = acc03[r];
        }
        if (crow1 + r < n_nodes) {
            op1[(long)r * F_OUT +  0] = acc10[r];
            op1[(long)r * F_OUT + 16] = acc11[r];
            op1[(long)r * F_OUT + 32] = acc12[r];
            op1[(long)r * F_OUT + 48] = acc13[r];
        }
    }
}

// ---------------------------------------------------------------------------
// SpMM: out += A_coo @ support. Row-sorted edges -> register-accumulated
// segmented reduction, atomic flush only on row change / chunk boundary.
// Edge metadata staged through LDS. support (102.4 MB) is L2-resident
// (192 MB global L2), so the 3.3 GB gather stream is L2 bandwidth.
// ---------------------------------------------------------------------------
__global__ __launch_bounds__(64)
void spmm_coo(const float* __restrict__ support, const int* __restrict__ erow,
              const int* __restrict__ ecol, const float* __restrict__ eval,
              float* __restrict__ out, int n_edges) {
    __shared__ int   s_row[EPB];
    __shared__ int   s_col[EPB];
    __shared__ float s_val[EPB];

    const int t  = threadIdx.x;                 // 0..63
    const int e0 = blockIdx.x * EPB;
    int cnt = n_edges - e0;
    if (cnt > EPB) cnt = EPB;
    if (cnt <= 0) return;

    for (int i = t; i < cnt; i += 64) {
        s_row[i] = erow[e0 + i];
        s_col[i] = ecol[e0 + i];
        s_val[i] = eval[e0 + i];
    }
    __syncthreads();

    const int c = t * 4;                        // this thread's 4 columns
    float4 acc = {0.f, 0.f, 0.f, 0.f};
    int cur = s_row[0];

    for (int i = 0; i < cnt; ++i) {
        const int r = s_row[i];
        if (r != cur) {
            float* o = out + (long)cur * F_OUT + c;
            atomicAdd(o + 0, acc.x); atomicAdd(o + 1, acc.y);
            atomicAdd(o + 2, acc.z); atomicAdd(o + 3, acc.w);
            acc = {0.f, 0.f, 0.f, 0.f};
            cur = r;
        }
        const float v = s_val[i];
        const float4 s = *(const float4*)(support + (long)s_col[i] * F_OUT + c);
        acc.x += v * s.x; acc.y += v * s.y; acc.z += v * s.z; acc.w += v * s.w;
    }
    float* o = out + (long)cur * F_OUT + c;
    atomicAdd(o + 0, acc.x); atomicAdd(o + 1, acc.y);
    atomicAdd(o + 2, acc.z); atomicAdd(o + 3, acc.w);
}

// ---------------------------------------------------------------------------
extern "C" void kernel_launch(void* const* d_in, const int* in_sizes, int n_in,
                              void* d_out, int out_size, void* d_ws, size_t ws_size,
                              hipStream_t stream) {
    const float* x    = (const float*)d_in[0];   // [N, 512]
    const float* w    = (const float*)d_in[1];   // [512, 256]
    const int*   erow = (const int*)  d_in[2];   // [E]
    const int*   ecol = (const int*)  d_in[3];   // [E]
    const float* eval = (const float*)d_in[4];   // [E]
    float*       out  = (float*)d_out;           // [N, 256]

    const int n_nodes = in_sizes[0] / F_IN;
    const int n_edges = in_sizes[2];

    float* support = (float*)d_ws;               // [N, 256] scratch

    // 1) zero output (poisoned by harness)
    const long n4 = (long)out_size / 4;
    zero_f4<<<dim3((unsigned)((n4 + 255) / 256)), dim3(256), 0, stream>>>((float4*)out, n4);

    // 2) dense GEMM via fp32 WMMA + TDM double-buffered LDS staging
    gemm_xw_wmma<<<dim3((unsigned)((n_nodes + 63) / 64)), dim3(256), 0, stream>>>(
        x, w, support, n_nodes);

    // 3) sparse scatter-add (segmented reduction over row-sorted COO)
    spmm_coo<<<dim3((unsigned)((n_edges + EPB - 1) / EPB)), dim3(64), 0, stream>>>(
        support, erow, ecol, eval, out, n_edges);
}